// Quant3Linear_3401614098714
// MI455X (gfx1250) — compile-verified
//
#include <hip/hip_runtime.h>
#include <hip/hip_bf16.h>

// ---------------------------------------------------------------------------
// QuIP-style 4-bit quantized linear, single token, for MI455X (gfx1250).
//   out = butterflyU( W_deq @ butterflyV(x/scaleWH) ) + bias
// Memory-bound on the 256MB qweight stream (HBM floor ~11.5us @ 23.3TB/s).
// GEMV is expressed through V_WMMA_F32_16X16X4_F32 (full fp32 precision).
// ---------------------------------------------------------------------------

#define IN_F   8192
#define OUT_F  8192
#define LOG2N  13
#define HALF_N 4096
#define SPLITK 16
#define KCHUNK (IN_F / SPLITK)   // 512 rows of qweight per split

typedef __attribute__((ext_vector_type(2))) float v2f;
typedef __attribute__((ext_vector_type(8))) float v8f;

// ---------------------------------------------------------------------------
// Kernel A: h = butterflyV(x / scaleWH);  also S = sum(h).
// Single workgroup, 1024 threads, data lives in LDS (32KB).
// ---------------------------------------------------------------------------
__global__ __launch_bounds__(1024) void butterfly_v_kernel(
    const float* __restrict__ x, const float* __restrict__ sWH,
    const float* __restrict__ ang, const int* __restrict__ p_in,
    const int* __restrict__ p_out, float* __restrict__ h_out,
    float* __restrict__ S_out) {
  __shared__ float buf[IN_F];
  __shared__ float red[1024];
  const int tid = threadIdx.x;

  // input permutation + diagonal scaling: buf[k] = (x/sWH)[p_in[k]]
  for (int k = tid; k < IN_F; k += 1024) {
    int pi = p_in[k];
    buf[k] = x[pi] / sWH[pi];
  }
  __syncthreads();

  // 13 stages: pair m -> (c*a + s*b) at m, (c*b - s*a) at HALF_N+m
  for (int s = 0; s < LOG2N; ++s) {
    float r0[4], r1[4];
#pragma unroll
    for (int t = 0; t < 4; ++t) {
      int j = tid + t * 1024;            // pair index 0..4095
      float a = buf[2 * j];
      float b = buf[2 * j + 1];
      float th = ang[s * HALF_N + j];
      float c = cosf(th), sn = sinf(th);
      r0[t] = c * a + sn * b;
      r1[t] = c * b - sn * a;
    }
    __syncthreads();
#pragma unroll
    for (int t = 0; t < 4; ++t) {
      int j = tid + t * 1024;
      buf[j] = r0[t];
      buf[HALF_N + j] = r1[t];
    }
    __syncthreads();
  }

  // output permutation + running sum
  float local = 0.f;
  for (int k = tid; k < IN_F; k += 1024) {
    float v = buf[p_out[k]];
    h_out[k] = v;
    local += v;
  }
  red[tid] = local;
  __syncthreads();
  for (int off = 512; off > 0; off >>= 1) {
    if (tid < off) red[tid] += red[tid + off];
    __syncthreads();
  }
  if (tid == 0) S_out[0] = red[0];
}

// ---------------------------------------------------------------------------
// Kernel B: split-K GEMV via WMMA f32 16x16x4.
//   part[split][o] = sum_{i in split-chunk} q[i][o] * h[i]
// grid = (OUT_F/128, SPLITK), block = 256 (8 waves); each wave owns one
// 16-output tile and streams a 512-row K slice.
// A (weights, documented layout): lane m holds M=m; v0 = {K0 | K2 upper half},
// v1 = {K1 | K3}. B (h broadcast across N-lanes): v0 = {h[K0] | h[K2]},
// v1 = {h[K1] | h[K3]} — lane-half K offset = Kspan/2, matching the
// documented CDNA5 WMMA f32 A 16x4 operand layout.
// All D columns are identical; lanes 0 and 16 hold M=0..7 / M=8..15.
// ---------------------------------------------------------------------------
__global__ __launch_bounds__(256) void gemv_wmma_kernel(
    const int* __restrict__ q, const float* __restrict__ h,
    float* __restrict__ part) {
  __shared__ float lds_h[KCHUNK];
  const int tid  = threadIdx.x;
  const int wave = tid >> 5;
  const int lane = tid & 31;
  const int half = lane >> 4;     // 0: lanes 0-15, 1: lanes 16-31
  const int m    = lane & 15;     // output row within tile / N column
  const int i0   = blockIdx.y * KCHUNK;
  const int o    = (blockIdx.x * 8 + wave) * 16;

  // stage the h slice for this K chunk into LDS (broadcast reads later)
  for (int t = tid; t < KCHUNK; t += 256) lds_h[t] = h[i0 + t];
  __syncthreads();

  v8f acc = {0.f, 0.f, 0.f, 0.f, 0.f, 0.f, 0.f, 0.f};
  const size_t colbase = (size_t)o + (size_t)m;

#pragma unroll 4
  for (int i = 0; i < KCHUNK; i += 4) {
    const int j = i + 2 * half;                       // this lane-half's K base
    const int* p = q + (size_t)(i0 + j) * OUT_F + colbase;
    // rows j and j+1 of qweight, 16 consecutive outputs per half-wave.
    // Non-temporal: 256MB single-use stream, keep it out of L2.
    int w0 = __builtin_nontemporal_load(p);
    int w1 = __builtin_nontemporal_load(p + OUT_F);
    v2f A;
    A.x = (float)w0;          // K = 0 + 2*half
    A.y = (float)w1;          // K = 1 + 2*half
    float2 hb = *reinterpret_cast<const float2*>(&lds_h[j]);  // ds_load_b64
    v2f B;
    B.x = hb.x;               // h[K = 0 + 2*half], broadcast across N lanes
    B.y = hb.y;               // h[K = 1 + 2*half]
    // D = A x B + C  (8 args: neg_a, A, neg_b, B, c_mod, C, reuse_a, reuse_b)
    acc = __builtin_amdgcn_wmma_f32_16x16x4_f32(false, A, false, B,
                                                (short)0, acc, false, false);
  }

  // every N column equal; lane 0 carries M=0..7 (acc[0..7]), lane 16 M=8..15.
  if (m == 0) {
    float* dst = part + (size_t)blockIdx.y * OUT_F + (size_t)o + half * 8;
#pragma unroll
    for (int r = 0; r < 8; ++r) dst[r] = acc[r];
  }
}

// ---------------------------------------------------------------------------
// Kernel C: reduce split-K partials, apply affine dequant correction,
// U-side butterfly, add bias.  g[o] = (2*scale/15)*acc[o] - scale*S
// ---------------------------------------------------------------------------
__global__ __launch_bounds__(1024) void butterfly_u_kernel(
    const float* __restrict__ part, const float* __restrict__ scale_p,
    const float* __restrict__ S_p, const float* __restrict__ bias,
    const float* __restrict__ ang, const int* __restrict__ p_in,
    const int* __restrict__ p_out, float* __restrict__ out) {
  __shared__ float buf[OUT_F];
  const int tid = threadIdx.x;
  const float scale = scale_p[0];
  const float a  = 2.0f * scale / 15.0f;
  const float c0 = -scale * S_p[0];

  // split-K reduction fused with input permutation + affine correction
  for (int k = tid; k < OUT_F; k += 1024) {
    int oidx = p_in[k];
    float accv = 0.f;
#pragma unroll
    for (int s = 0; s < SPLITK; ++s) accv += part[s * OUT_F + oidx];
    buf[k] = a * accv + c0;
  }
  __syncthreads();

  for (int s = 0; s < LOG2N; ++s) {
    float r0[4], r1[4];
#pragma unroll
    for (int t = 0; t < 4; ++t) {
      int j = tid + t * 1024;
      float av = buf[2 * j];
      float bv = buf[2 * j + 1];
      float th = ang[s * HALF_N + j];
      float c = cosf(th), sn = sinf(th);
      r0[t] = c * av + sn * bv;
      r1[t] = c * bv - sn * av;
    }
    __syncthreads();
#pragma unroll
    for (int t = 0; t < 4; ++t) {
      int j = tid + t * 1024;
      buf[j] = r0[t];
      buf[HALF_N + j] = r1[t];
    }
    __syncthreads();
  }

  for (int k = tid; k < OUT_F; k += 1024) out[k] = buf[p_out[k]] + bias[k];
}

// ---------------------------------------------------------------------------
// Workspace layout (floats): [0,8192) h; [8192] S; [8448, 8448+16*8192) partials
// ---------------------------------------------------------------------------
extern "C" void kernel_launch(void* const* d_in, const int* in_sizes, int n_in,
                              void* d_out, int out_size, void* d_ws, size_t ws_size,
                              hipStream_t stream) {
  const float* x      = (const float*)d_in[0];
  const int*   qw     = (const int*)d_in[1];
  const float* sWH    = (const float*)d_in[2];
  const float* scale  = (const float*)d_in[3];
  const float* bias   = (const float*)d_in[4];
  const float* U_ang  = (const float*)d_in[5];
  const float* V_ang  = (const float*)d_in[6];
  const int*   pU_in  = (const int*)d_in[7];
  const int*   pU_out = (const int*)d_in[8];
  const int*   pV_in  = (const int*)d_in[9];
  const int*   pV_out = (const int*)d_in[10];

  float* ws   = (float*)d_ws;
  float* h    = ws;                  // 8192
  float* S    = ws + IN_F;           // 1
  float* part = ws + IN_F + 256;     // SPLITK * OUT_F

  butterfly_v_kernel<<<1, 1024, 0, stream>>>(x, sWH, V_ang, pV_in, pV_out, h, S);
  gemv_wmma_kernel<<<dim3(OUT_F / 128, SPLITK), 256, 0, stream>>>(qw, h, part);
  butterfly_u_kernel<<<1, 1024, 0, stream>>>(part, scale, S, bias, U_ang,
                                             pU_in, pU_out, (float*)d_out);
}